// BatchNormNodes_8821862826757
// MI455X (gfx1250) — compile-verified
//
#include <hip/hip_runtime.h>
#include <hip/hip_bf16.h>

#define H_DIM 256
#define N_DIM 256
#define B_DIM 4
#define ROWS  (B_DIM * N_DIM)   // 1024
#define NPART 32                // stats partial blocks

typedef float v2f __attribute__((ext_vector_type(2)));
typedef float v4f __attribute__((ext_vector_type(4)));
typedef float v8f __attribute__((ext_vector_type(8)));

// sigmoid: prefer single-transcendental v_tanh_f32 path on gfx1250
#if __has_builtin(__builtin_amdgcn_tanhf)
__device__ __forceinline__ float fast_sigmoid(float x) {
  return fmaf(0.5f, __builtin_amdgcn_tanhf(0.5f * x), 0.5f);
}
#else
__device__ __forceinline__ float fast_sigmoid(float x) {
  return __builtin_amdgcn_rcpf(1.0f + __expf(-x));
}
#endif

// ---------------------------------------------------------------------------
// Kernel 1: x_left = nodes @ W1^T, x_w2 = nodes @ W2^T  (fp32 WMMA 16x16x4)
// One wave computes one 16x16 tile of BOTH outputs (A fragment reused).
// ---------------------------------------------------------------------------
__global__ void proj_wmma_kernel(const float* __restrict__ x,
                                 const float* __restrict__ W1,
                                 const float* __restrict__ W2,
                                 float* __restrict__ y1,
                                 float* __restrict__ y2) {
  const int wave = (blockIdx.x * blockDim.x + threadIdx.x) >> 5; // global wave
  const int lane = threadIdx.x & 31;
  const int mt = wave >> 4;          // 64 row tiles
  const int nt = wave & 15;          // 16 col tiles
  const int m0 = mt * 16;
  const int n0 = nt * 16;
  const int half = lane >> 4;        // 0: K {0,1}; 1: K {2,3}
  const int l = lane & 15;

  const float* xrow  = x  + (size_t)(m0 + l) * H_DIM;
  const float* w1row = W1 + (size_t)(n0 + l) * H_DIM;
  const float* w2row = W2 + (size_t)(n0 + l) * H_DIM;

  v8f c1 = {};
  v8f c2 = {};

  #pragma unroll 4
  for (int k0 = 0; k0 < H_DIM; k0 += 4) {
    const int k = k0 + half * 2;
    v2f a  = *(const v2f*)(xrow  + k);
    v2f b1 = *(const v2f*)(w1row + k);
    v2f b2 = *(const v2f*)(w2row + k);
    c1 = __builtin_amdgcn_wmma_f32_16x16x4_f32(false, a, false, b1,
                                               (short)0, c1, false, false);
    c2 = __builtin_amdgcn_wmma_f32_16x16x4_f32(false, a, false, b2,
                                               (short)0, c2, false, false);
  }

  const int mbase = m0 + half * 8;
  #pragma unroll
  for (int r = 0; r < 8; ++r) {
    y1[(size_t)(mbase + r) * H_DIM + n0 + l] = c1[r];
    y2[(size_t)(mbase + r) * H_DIM + n0 + l] = c2[r];
  }
}

// ---------------------------------------------------------------------------
// Kernel 2: fused sigmoid + sum-over-j + weighted aggregation + add x_left.
// right[b,i,h] = (sum_j sig(e[b,i,j,h]) * xw2[b,j,h]) / (sum_j sig + eps)
// One block per (b,i). 256 threads = 64 channel-groups (float4 of h) x
// 4 j-chunks of 64. Non-temporal b128 streaming loads for the read-once
// edges tensor (don't thrash L2); cached loads for the reused xw2.
// ---------------------------------------------------------------------------
__global__ void aggregate_kernel(const float* __restrict__ edges,
                                 const float* __restrict__ xw2,
                                 const float* __restrict__ xleft,
                                 float* __restrict__ equ) {
  const int bi = blockIdx.x;              // b*N + i
  const int b  = bi >> 8;                 // bi / N
  const int t  = threadIdx.x;
  const int hg = (t & 63) << 2;           // channel group base: 0,4,...,252
  const int jc = t >> 6;                  // j-chunk 0..3 (64 j's each)

  const float* ebase = edges + (size_t)bi * N_DIM * H_DIM + hg;
  const float* xbase = xw2   + (size_t)b  * N_DIM * H_DIM + hg;

  v4f num = {0.f, 0.f, 0.f, 0.f};
  v4f den = {0.f, 0.f, 0.f, 0.f};

  const int j0 = jc * 64;
  #pragma unroll 4
  for (int j = j0; j < j0 + 64; ++j) {
    const v4f e  = __builtin_nontemporal_load(
                       (const v4f*)(ebase + (size_t)j * H_DIM));
    const v4f xv = *(const v4f*)(xbase + (size_t)j * H_DIM);
    const float s0 = fast_sigmoid(e.x);
    const float s1 = fast_sigmoid(e.y);
    const float s2 = fast_sigmoid(e.z);
    const float s3 = fast_sigmoid(e.w);
    num.x = fmaf(s0, xv.x, num.x);  den.x += s0;
    num.y = fmaf(s1, xv.y, num.y);  den.y += s1;
    num.z = fmaf(s2, xv.z, num.z);  den.z += s2;
    num.w = fmaf(s3, xv.w, num.w);  den.w += s3;
  }

  __shared__ v4f rnum[256];
  __shared__ v4f rden[256];
  rnum[t] = num;
  rden[t] = den;
  __syncthreads();

  if (jc == 0) {
    #pragma unroll
    for (int p = 1; p < 4; ++p) {
      num += rnum[t + p * 64];
      den += rden[t + p * 64];
    }
    const v4f xl = *(const v4f*)(xleft + (size_t)bi * H_DIM + hg);
    v4f o;
    o.x = xl.x + num.x / (den.x + 1e-20f);   // precise div, once per channel
    o.y = xl.y + num.y / (den.y + 1e-20f);
    o.z = xl.z + num.z / (den.z + 1e-20f);
    o.w = xl.w + num.w / (den.w + 1e-20f);
    *(v4f*)(equ + (size_t)bi * H_DIM + hg) = o;
  }
}

// ---------------------------------------------------------------------------
// Kernel 3a: partial batch-norm sums. Block p covers ROWS/NPART rows;
// thread h owns channel h (coalesced). Writes psum/psq[p][h].
// ---------------------------------------------------------------------------
__global__ void stats_partial_kernel(const float* __restrict__ equ,
                                     float* __restrict__ psum,
                                     float* __restrict__ psq) {
  const int p = blockIdx.x;
  const int h = threadIdx.x;
  const int r0 = p * (ROWS / NPART);
  float sum = 0.0f, sq = 0.0f;
  #pragma unroll 8
  for (int r = r0; r < r0 + ROWS / NPART; ++r) {
    const float v = equ[(size_t)r * H_DIM + h];
    sum += v;
    sq = fmaf(v, v, sq);
  }
  psum[p * H_DIM + h] = sum;
  psq[p * H_DIM + h] = sq;
}

// ---------------------------------------------------------------------------
// Kernel 3b: combine partials -> scale[h] = gamma*rsqrt(var+eps),
//            shift[h] = beta - mean*scale. (biased var, matches reference)
// ---------------------------------------------------------------------------
__global__ void stats_finalize_kernel(const float* __restrict__ psum,
                                      const float* __restrict__ psq,
                                      const float* __restrict__ gamma,
                                      const float* __restrict__ beta,
                                      float* __restrict__ scale,
                                      float* __restrict__ shift) {
  const int h = threadIdx.x;
  float sum = 0.0f, sq = 0.0f;
  #pragma unroll
  for (int p = 0; p < NPART; ++p) {
    sum += psum[p * H_DIM + h];
    sq  += psq[p * H_DIM + h];
  }
  const float inv_n = 1.0f / (float)ROWS;
  const float mean = sum * inv_n;
  const float var  = sq * inv_n - mean * mean;
  const float rstd = rsqrtf(var + 1e-5f);
  const float sc   = gamma[h] * rstd;
  scale[h] = sc;
  shift[h] = fmaf(-mean, sc, beta[h]);
}

// ---------------------------------------------------------------------------
// Kernel 4: out = equ * scale[h] + shift[h]   (float4 vectorized)
// ---------------------------------------------------------------------------
__global__ void normalize_kernel(const float* __restrict__ equ,
                                 const float* __restrict__ scale,
                                 const float* __restrict__ shift,
                                 float* __restrict__ out) {
  const int idx = blockIdx.x * blockDim.x + threadIdx.x;   // float4 index
  const int hg = idx & (H_DIM / 4 - 1);                    // float4 within row
  const v4f e  = ((const v4f*)equ)[idx];
  const v4f sc = ((const v4f*)scale)[hg];
  const v4f sh = ((const v4f*)shift)[hg];
  v4f o;
  o.x = fmaf(e.x, sc.x, sh.x);
  o.y = fmaf(e.y, sc.y, sh.y);
  o.z = fmaf(e.z, sc.z, sh.z);
  o.w = fmaf(e.w, sc.w, sh.w);
  ((v4f*)out)[idx] = o;
}

extern "C" void kernel_launch(void* const* d_in, const int* in_sizes, int n_in,
                              void* d_out, int out_size, void* d_ws, size_t ws_size,
                              hipStream_t stream) {
  const float* nodes = (const float*)d_in[0];  // (B,N,H)      262144
  const float* edges = (const float*)d_in[1];  // (B,N,N,H)  67108864
  const float* W1    = (const float*)d_in[2];  // (H,H)         65536
  const float* W2    = (const float*)d_in[3];  // (H,H)         65536
  const float* gamma = (const float*)d_in[4];  // (H,)            256
  const float* beta  = (const float*)d_in[5];  // (H,)            256
  float* out = (float*)d_out;                  // (B,N,H)      262144

  // Workspace layout (floats)
  float* ws    = (float*)d_ws;
  float* xleft = ws;                               // ROWS*H
  float* xw2   = xleft + (size_t)ROWS * H_DIM;     // ROWS*H
  float* equ   = xw2   + (size_t)ROWS * H_DIM;     // ROWS*H
  float* psum  = equ   + (size_t)ROWS * H_DIM;     // NPART*H
  float* psq   = psum  + (size_t)NPART * H_DIM;    // NPART*H
  float* scale = psq   + (size_t)NPART * H_DIM;    // H
  float* shift = scale + H_DIM;                    // H

  // 1) projections: 1024 tiles, 4 waves (128 threads) per block -> 256 blocks
  proj_wmma_kernel<<<256, 128, 0, stream>>>(nodes, W1, W2, xleft, xw2);

  // 2) fused gating + aggregation: the single pass over the 268 MB edges
  aggregate_kernel<<<ROWS, 256, 0, stream>>>(edges, xw2, xleft, equ);

  // 3) batch-norm statistics: parallel partials + tiny combine
  stats_partial_kernel<<<NPART, H_DIM, 0, stream>>>(equ, psum, psq);
  stats_finalize_kernel<<<1, H_DIM, 0, stream>>>(psum, psq, gamma, beta,
                                                 scale, shift);

  // 4) apply affine normalization (float4)
  normalize_kernel<<<(ROWS * H_DIM / 4) / 256, 256, 0, stream>>>(equ, scale,
                                                                 shift, out);
}